// MIX_48576080118410
// MI455X (gfx1250) — compile-verified
//
#include <hip/hip_runtime.h>
#include <math.h>

// MI455X / gfx1250 kernel.
// Memory-bound mixture-of-activations kernel (~804 MB traffic -> ~35us at 23.3 TB/s).
// V_WMMA_F32_16X16X4_F32 computes the per-neuron 4x4 logits GEMM (K=4 == WMMA K).
// Raw logits are staged through LDS so softmax + weighted-sum run in an
// elementwise phase with all 32 lanes live (the WMMA D-layout only populates
// 8/32 lanes, so reducing there is 4x wasteful), and the dominant alpha stream
// (536 MB) is written as fully coalesced float4 stores. No IEEE division:
// v_rcp_f32 / v_exp_f32 / v_tanh_f32 only.

typedef __attribute__((ext_vector_type(2))) float v2f;
typedef __attribute__((ext_vector_type(4))) float f4;
typedef __attribute__((ext_vector_type(8))) float v8f;

#define BT 32          // batch rows per block tile
#define NT 64          // neurons per block tile
#define SP 68          // padded s-tile row stride (floats): 272B/row, 16B aligned, bank-shift 4/row
#define AP 260         // logit staging row stride (floats): 1040B/row, 16B aligned; rows r,r+8 on different banks

__device__ __forceinline__ float fast_rcp(float x) {
    return __builtin_amdgcn_rcpf(x);          // v_rcp_f32
}

__device__ __forceinline__ float fast_sigmoid(float x) {
    return fast_rcp(1.0f + __expf(-x));       // v_exp_f32 + v_rcp_f32
}

__device__ __forceinline__ float fast_tanh(float x) {
#if __has_builtin(__builtin_amdgcn_tanhf)
    return __builtin_amdgcn_tanhf(x);         // v_tanh_f32 (CDNA5 transcendental)
#else
    float t = __expf(-2.0f * x);
    return (1.0f - t) * fast_rcp(1.0f + t);
#endif
}

__global__ void __launch_bounds__(256)
act_attn_wmma_kernel(const float* __restrict__ s,
                     const float* __restrict__ W,
                     const float* __restrict__ bias,
                     float* __restrict__ res_out,
                     float* __restrict__ alpha_out,
                     int N)
{
    __shared__ float sT[BT * SP];        // s tile, [row][n], padded
    __shared__ float wT[NT * 16];        // W tile, [n][k][l]
    __shared__ float bT[NT * 4];         // bias tile, [n][l]
    __shared__ float aT[BT * AP];        // raw-logit staging, [row][n*4 + l], padded

    const int t  = threadIdx.x;
    const int nb = N / NT;
    const int n0 = (blockIdx.x % nb) * NT;
    const int b0 = (blockIdx.x / nb) * BT;

    // ---- Phase 0: coalesced global -> LDS staging ------------------------
    {
        // s tile: BT x NT floats = 512 float4, 2 per thread
#pragma unroll
        for (int i = 0; i < 2; ++i) {
            int idx = t + i * 256;              // 0..511 float4 slots
            int row = idx >> 4;                 // 16 float4 per row
            int c4  = idx & 15;
            const f4* srow = (const f4*)(s + (size_t)(b0 + row) * N + n0);
            *(f4*)&sT[row * SP + c4 * 4] = srow[c4];
        }
        // W tile: NT*16 = 1024 floats = 256 float4, 1 per thread (contiguous)
        const f4* w4 = (const f4*)(W + (size_t)n0 * 16);
        *(f4*)&wT[t * 4] = w4[t];
        // bias tile: NT*4 = 256 floats, 1 per thread (contiguous)
        bT[t] = bias[(size_t)n0 * 4 + t];
    }
    __syncthreads();

    // ---- Phase 1: WMMA logits (C = inline 0) -> LDS staging --------------
    const int wave = t >> 5;
    const int lane = t & 31;
    const int col  = lane & 15;       // D/B column index (only 0..3 carry data)
    const int half = lane >> 4;       // 0: K rows 0/1, 1: K rows 2/3
    const int c3   = col & 3;
    const bool live = (col < 4);

#pragma unroll
    for (int q = 0; q < 8; ++q) {
        const int nl = wave * 8 + q;          // local neuron index for this wave

        // B operand: W[n] in columns 0..3, zero elsewhere.
        // Layout: VGPR0 = K row (2*half), VGPR1 = K row (2*half+1), N = col.
        float bx = wT[nl * 16 + (half * 2 + 0) * 4 + c3];
        float by = wT[nl * 16 + (half * 2 + 1) * 4 + c3];
        bx = live ? bx : 0.0f;
        by = live ? by : 0.0f;
        v2f Bm = {bx, by};

        // A operands: activation rows. Lane m (0-15) carries K=0,1 of batch row m;
        // lane 16+m carries K=2,3 of batch row m. Acts: [relu, sigmoid, tanh, id].
        float sA = sT[(lane & 15) * SP + nl];         // rows 0..15
        float sB = sT[(16 + (lane & 15)) * SP + nl];  // rows 16..31
        v2f A1 = { half ? fast_tanh(sA) : fmaxf(sA, 0.0f),
                   half ? sA            : fast_sigmoid(sA) };
        v2f A2 = { half ? fast_tanh(sB) : fmaxf(sB, 0.0f),
                   half ? sB            : fast_sigmoid(sB) };

        // logits[16x16]: D[m][l] = sum_k acts[m][k] * W[n][k][l]  (bias added in phase 2)
        v8f C = {};                                     // inline 0 SRC2
        v8f d1 = __builtin_amdgcn_wmma_f32_16x16x4_f32(
            false, A1, false, Bm, (short)0, C, false, false);
        v8f d2 = __builtin_amdgcn_wmma_f32_16x16x4_f32(
            false, A2, false, Bm, (short)0, C, false, false);

        // Stage raw logits: D reg v, lanes 0-3 -> row v; lanes 16-19 -> row v+8.
        if (live) {
#pragma unroll
            for (int v = 0; v < 8; ++v) {
                int r1 = v + half * 8;
                aT[r1 * AP + nl * 4 + c3]        = d1[v];
                aT[(r1 + 16) * AP + nl * 4 + c3] = d2[v];
            }
        }
    }
    __syncthreads();

    // ---- Phase 2: elementwise softmax + weighted sum, coalesced stores ---
    // All 32 lanes live: lane = neuron (fastest dim), float4 alpha per thread.
#pragma unroll
    for (int p = 0; p < 8; ++p) {
        int bl = p * 4 + (t >> 6);
        int nl = t & 63;

        f4 lg = *(const f4*)&aT[bl * AP + nl * 4];  // raw logits
        f4 bs = *(const f4*)&bT[nl * 4];            // bias
        lg.x += bs.x; lg.y += bs.y; lg.z += bs.z; lg.w += bs.w;

        // softmax over K=4 (max-subtracted, rcp instead of division)
        float m = fmaxf(fmaxf(lg.x, lg.y), fmaxf(lg.z, lg.w));
        f4 e;
        e.x = __expf(lg.x - m);
        e.y = __expf(lg.y - m);
        e.z = __expf(lg.z - m);
        e.w = __expf(lg.w - m);
        float rs = fast_rcp(e.x + e.y + e.z + e.w);
        f4 a = { e.x * rs, e.y * rs, e.z * rs, e.w * rs };

        // res = sum_l alpha[l] * act[l]
        float sv = sT[bl * SP + nl];
        float r = a.x * fmaxf(sv, 0.0f)
                + a.y * fast_sigmoid(sv)
                + a.z * fast_tanh(sv)
                + a.w * sv;

        size_t g = (size_t)(b0 + bl) * N + (n0 + nl);
        res_out[g] = r;
        *(f4*)&alpha_out[g * 4] = a;
    }
}

extern "C" void kernel_launch(void* const* d_in, const int* in_sizes, int n_in,
                              void* d_out, int out_size, void* d_ws, size_t ws_size,
                              hipStream_t stream) {
    const float* s    = (const float*)d_in[0];   // [B, N]
    const float* W    = (const float*)d_in[1];   // [N, K, K], K=4
    const float* bias = (const float*)d_in[2];   // [N, K]

    const int N = in_sizes[1] / 16;              // N*K*K / 16
    const int B = in_sizes[0] / N;

    float* res_out   = (float*)d_out;            // [B, N]
    float* alpha_out = res_out + (size_t)B * N;  // [B, N, 4]

    const int grid = (B / BT) * (N / NT);
    act_attn_wmma_kernel<<<grid, 256, 0, stream>>>(s, W, bias, res_out, alpha_out, N);
}